// CorrelationModule_55430847922401
// MI455X (gfx1250) — compile-verified
//
#include <hip/hip_runtime.h>
#include <math.h>
#include <stdint.h>

typedef __attribute__((ext_vector_type(2))) float v2f;
typedef __attribute__((ext_vector_type(8))) float v8f;

#define BB   2
#define CC   256
#define NN   64
#define KCLS 10
#define PITCH 68   // LDS row pitch (floats): 68%64==4 -> conflict-free strided rows

// ---------------------------------------------------------------------------
// gfx1250 async global->LDS staging (ASYNCcnt path), with portable fallback.
// Builtin signature (from hipcc diagnostic): (int4 AS1*, int4 AS3*, imm, imm)
// ---------------------------------------------------------------------------
#if defined(__has_builtin)
#if __has_builtin(__builtin_amdgcn_global_load_async_to_lds_b128)
#define HAVE_ASYNC_LDS 1
#endif
#endif

#ifdef HAVE_ASYNC_LDS
typedef int v4i __attribute__((vector_size(16)));
typedef __attribute__((address_space(1))) v4i GV4;   // global int4
typedef __attribute__((address_space(3))) v4i LV4;   // LDS int4
__device__ __forceinline__ void async_g2l_b128(const float* g, float* l) {
    // generic->AS casts via uintptr_t: AS1 keeps the 64-bit address; AS3 ptr is
    // 32-bit and the low 32 bits of a generic LDS address are the LDS offset.
    __builtin_amdgcn_global_load_async_to_lds_b128(
        (GV4*)(uintptr_t)g, (LV4*)(uintptr_t)l, 0, 0);
}
__device__ __forceinline__ void wait_async_lds() {
#if __has_builtin(__builtin_amdgcn_s_wait_asynccnt)
    __builtin_amdgcn_s_wait_asynccnt(0);
#else
    asm volatile("s_wait_asynccnt 0x0" ::: "memory");
#endif
}
#endif

// Stage ROWS x 64 floats from global (row stride gstride) into LDS (pitch 68).
template <int ROWS>
__device__ __forceinline__ void stage64(const float* __restrict__ g, size_t gstride,
                                        float* __restrict__ lds, int t) {
    #pragma unroll
    for (int e = t; e < ROWS * 16; e += 256) {
        const int r  = e >> 4;
        const int c4 = (e & 15) << 2;
#ifdef HAVE_ASYNC_LDS
        async_g2l_b128(g + (size_t)r * gstride + c4, lds + r * PITCH + c4);
#else
        const float4 v = *(const float4*)(g + (size_t)r * gstride + c4);
        *(float4*)(lds + r * PITCH + c4) = v;
#endif
    }
}

__device__ __forceinline__ void stage_fence() {
#ifdef HAVE_ASYNC_LDS
    wait_async_lds();
#endif
    __syncthreads();
}

// ---------------------------------------------------------------------------
// K1: normalized Gaussian heatmaps at level resolution.
// grid = (B*N, 3 levels), block = 256
// ---------------------------------------------------------------------------
__global__ void heat_kernel(const float* __restrict__ centers,
                            const int*   __restrict__ idx,
                            float* __restrict__ h0,
                            float* __restrict__ h1,
                            float* __restrict__ h2) {
    const int bn    = blockIdx.x;      // b*64+n
    const int level = blockIdx.y;      // 0,1,2
    const int W     = 64 >> level;
    const int HW    = W * W;
    const int scale = 8 << level;      // 512/W
    float* heat = (level == 0 ? h0 : (level == 1 ? h1 : h2)) + (size_t)bn * HW;

    const int  t       = threadIdx.x;
    const bool invalid = (idx[bn] == -1);
    const float cx = centers[bn * 2 + 0];
    const float cy = centers[bn * 2 + 1];
    const float alpha = -0.5f / (3.0f * 3.0f);

    float vals[16];
    const int per = HW >> 8;           // 16 / 4 / 1
    float lsum = 0.0f;
    for (int i = 0; i < per; ++i) {
        const int e  = t + (i << 8);
        const int hh = e / W;
        const int ww = e - hh * W;
        const float x  = (float)(ww * scale) + 0.5f;
        const float y  = (float)(hh * scale) + 0.5f;
        const float dx = x - cx, dy = y - cy;
        const float v  = invalid ? 0.0f : expf(alpha * (dx * dx + dy * dy));
        vals[i] = v;
        lsum += v;
    }
    __shared__ float red[256];
    red[t] = lsum;
    __syncthreads();
    for (int s = 128; s > 0; s >>= 1) {
        if (t < s) red[t] += red[t + s];
        __syncthreads();
    }
    const float inv = 1.0f / (red[0] + 1e-8f);
    for (int i = 0; i < per; ++i) heat[t + (i << 8)] = vals[i] * inv;
}

// ---------------------------------------------------------------------------
// K2: template GEMM  tmpl[n,c] = sum_hw heat[n,hw] * feat[c,hw]
// M=n (64 -> 4 tiles), N=c (256 -> 8 groups of 32), K=hw.
// grid = (8 c-groups, B), block = 256 (8 waves: wave = ntile + 4*csub)
// ---------------------------------------------------------------------------
__global__ void tmpl_kernel(const float* __restrict__ feat,
                            const float* __restrict__ heat,
                            float* __restrict__ tmpl,
                            int HW) {
    const int b  = blockIdx.y;
    const int cg = blockIdx.x;                 // c-group: 32 channels
    const float* fb = feat + (size_t)b * CC * HW + (size_t)cg * 32 * HW;
    const float* hb = heat + (size_t)b * NN * HW;
    float*       tb = tmpl + (size_t)b * NN * CC;

    __shared__ float ldsH[64 * PITCH];
    __shared__ float ldsF[32 * PITCH];

    const int t      = threadIdx.x;
    const int lane   = t & 31;
    const int wv     = t >> 5;
    const int lane16 = lane & 15;
    const int khi    = lane >> 4;              // 0/1 -> K offset 0/2
    const int ntile  = wv & 3;
    const int csub   = wv >> 2;

    v8f acc = {};
    for (int k0 = 0; k0 < HW; k0 += 64) {
        stage64<64>(hb + k0, (size_t)HW, ldsH, t);
        stage64<32>(fb + k0, (size_t)HW, ldsF, t);
        stage_fence();

        const float* ar = &ldsH[(ntile * 16 + lane16) * PITCH];
        const float* br = &ldsF[(csub  * 16 + lane16) * PITCH];
        #pragma unroll
        for (int kk = 0; kk < 64; kk += 4) {
            const int kb = kk + khi * 2;
            v2f a, bmat;
            a[0]    = ar[kb];     a[1]    = ar[kb + 1];
            bmat[0] = br[kb];     bmat[1] = br[kb + 1];
            acc = __builtin_amdgcn_wmma_f32_16x16x4_f32(
                false, a, false, bmat, (short)0, acc, false, false);
        }
        __syncthreads();
    }
    // D layout: VGPR r -> M=r (lanes 0-15) / M=r+8 (lanes 16-31); N = lane16
    const int c_col = cg * 32 + csub * 16 + lane16;
    const int nbase = ntile * 16 + ((lane >= 16) ? 8 : 0);
    #pragma unroll
    for (int r = 0; r < 8; ++r) tb[(nbase + r) * CC + c_col] = acc[r];
}

// ---------------------------------------------------------------------------
// K3: correlation GEMM + class-wise max + [feat, cmax] concat.
// cor[n,hw] = sum_c tmpl[n,c]*feat[c,hw]; M=n(64), N=hw(32/block), K=C=256.
// grid = (HW/32, B), block = 256 (8 waves: wave = ntile + 4*hsub)
// ---------------------------------------------------------------------------
__global__ void corr_kernel(const float* __restrict__ feat,
                            const float* __restrict__ tmpl,
                            const int*   __restrict__ labels,
                            float* __restrict__ out,
                            int HW) {
    const int b   = blockIdx.y;
    const int hw0 = blockIdx.x * 32;
    const float* fb = feat + (size_t)b * CC * HW;
    const float* tb = tmpl + (size_t)b * NN * CC;
    float*       ob = out  + (size_t)b * (CC + KCLS) * HW;

    const int t      = threadIdx.x;
    const int lane   = t & 31;
    const int wv     = t >> 5;
    const int lane16 = lane & 15;
    const int khi    = lane >> 4;
    const int ntile  = wv & 3;
    const int hsub   = wv >> 2;

    __shared__ float ldsT[64 * PITCH];
    __shared__ float ldsC[64 * 36];

    // feat passthrough for this 32-wide hw slab (B128 copies)
    #pragma unroll
    for (int e = t; e < CC * 8; e += 256) {
        const int c  = e >> 3;
        const int c4 = (e & 7) << 2;
        const float4 v = *(const float4*)(fb + (size_t)c * HW + hw0 + c4);
        *(float4*)(ob + (size_t)c * HW + hw0 + c4) = v;
    }

    v8f acc = {};
    for (int k0 = 0; k0 < CC; k0 += 64) {
        stage64<64>(tb + k0, (size_t)CC, ldsT, t);
        stage_fence();

        const float* ar   = &ldsT[(ntile * 16 + lane16) * PITCH];
        const float* bptr = fb + (size_t)k0 * HW + hw0 + hsub * 16 + lane16;
        #pragma unroll
        for (int kk = 0; kk < 64; kk += 4) {
            const int kb = kk + khi * 2;
            v2f a, bmat;
            a[0]    = ar[kb];                     a[1]    = ar[kb + 1];
            bmat[0] = bptr[(size_t)kb * HW];      bmat[1] = bptr[(size_t)(kb + 1) * HW];
            acc = __builtin_amdgcn_wmma_f32_16x16x4_f32(
                false, a, false, bmat, (short)0, acc, false, false);
        }
        __syncthreads();
    }
    // dump cor tile to LDS: rows = n, cols = local hw
    const int col   = hsub * 16 + lane16;
    const int nbase = ntile * 16 + ((lane >= 16) ? 8 : 0);
    #pragma unroll
    for (int r = 0; r < 8; ++r) ldsC[(nbase + r) * 36 + col] = acc[r];
    __syncthreads();

    // class-wise max over n (labels only; invalid rows contribute cor==0)
    if (t < KCLS * 32) {
        const int k   = t >> 5;        // class index 0..9  (class id k+1)
        const int hl  = t & 31;
        const int cls = k + 1;
        float m = -3.402823466e38f;
        bool any = false;
        for (int n = 0; n < NN; ++n) {
            if (labels[b * NN + n] == cls) {
                any = true;
                const float v = ldsC[n * 36 + hl];
                m = (v > m) ? v : m;
            }
        }
        ob[(size_t)(CC + k) * HW + hw0 + hl] = any ? m : 0.0f;
    }
}

// ---------------------------------------------------------------------------
extern "C" void kernel_launch(void* const* d_in, const int* in_sizes, int n_in,
                              void* d_out, int out_size, void* d_ws, size_t ws_size,
                              hipStream_t stream) {
    const float* x0      = (const float*)d_in[0];  // [2,256,64,64]
    const float* x1      = (const float*)d_in[1];  // [2,256,32,32]
    const float* x2      = (const float*)d_in[2];  // [2,256,16,16]
    const float* centers = (const float*)d_in[3];  // [2,64,2]
    const int*   idx     = (const int*)d_in[4];    // [2,64]
    const int*   labels  = (const int*)d_in[5];    // [2,64]
    // d_in[6], d_in[7]: img_h, img_w == 512 (compile-time constants here)

    float* ws = (float*)d_ws;
    float* heat0 = ws;                                   // 2*64*4096
    float* heat1 = heat0 + (size_t)BB * NN * 4096;       // 2*64*1024
    float* heat2 = heat1 + (size_t)BB * NN * 1024;       // 2*64*256
    float* tmpl0 = heat2 + (size_t)BB * NN * 256;        // 2*64*256 each
    float* tmpl1 = tmpl0 + (size_t)BB * NN * CC;
    float* tmpl2 = tmpl1 + (size_t)BB * NN * CC;

    float* out0 = (float*)d_out;
    float* out1 = out0 + (size_t)BB * (CC + KCLS) * 4096;
    float* out2 = out1 + (size_t)BB * (CC + KCLS) * 1024;

    heat_kernel<<<dim3(BB * NN, 3), 256, 0, stream>>>(centers, idx, heat0, heat1, heat2);

    tmpl_kernel<<<dim3(8, BB), 256, 0, stream>>>(x0, heat0, tmpl0, 4096);
    tmpl_kernel<<<dim3(8, BB), 256, 0, stream>>>(x1, heat1, tmpl1, 1024);
    tmpl_kernel<<<dim3(8, BB), 256, 0, stream>>>(x2, heat2, tmpl2, 256);

    corr_kernel<<<dim3(4096 / 32, BB), 256, 0, stream>>>(x0, tmpl0, labels, out0, 4096);
    corr_kernel<<<dim3(1024 / 32, BB), 256, 0, stream>>>(x1, tmpl1, labels, out1, 1024);
    corr_kernel<<<dim3(256 / 32,  BB), 256, 0, stream>>>(x2, tmpl2, labels, out2, 256);
}